// RewardLoss_13151189860629
// MI455X (gfx1250) — compile-verified
//
#include <hip/hip_runtime.h>
#include <hip/hip_bf16.h>
#include <stdint.h>

#define NF    52
#define TB    256
#define DUMMY 42

// ---------------------------------------------------------------------------
// Node prep (FIRST in file so the disasm snippet shows the CDNA5 async path).
// Stage 256 node rows (52 f32 each) into LDS with async global->LDS copies:
// each wave stages its own 32 rows as 13 x global_load_async_to_lds_b128 per
// lane (512B per wave per issue), waits on ASYNCcnt, then does per-thread
// argmax scans out of LDS. Packs atom | (charge_idx<<8) | (arom<<16) |
// (max_bonds<<20) into one int, and counts nodes per graph.
// ---------------------------------------------------------------------------
__global__ void k_node_prep(const float* __restrict__ nf,
                            const float* __restrict__ mbt,
                            const int*   __restrict__ gid,
                            int*         __restrict__ meta,
                            float*       __restrict__ gcount,
                            int N) {
    __shared__ __align__(16) float tile[TB * NF];   // 52 KB
    const int  lane       = threadIdx.x & 31;
    const int  wave       = threadIdx.x >> 5;
    const int  blockStart = blockIdx.x * TB;
    const int  n          = blockStart + threadIdx.x;
    const bool full       = (blockStart + TB) <= N;

    if (full) {
        // Each wave asynchronously stages its 32 rows: 32*208B = 416 x 16B chunks.
        unsigned ldsBase = (unsigned)(uintptr_t)(&tile[wave * 32 * NF]);
        unsigned gByte   = (unsigned)(blockStart + wave * 32) * (NF * 4u);
#pragma unroll
        for (int it = 0; it < 13; ++it) {
            unsigned c  = (unsigned)(it * 32 + lane);
            unsigned lo = ldsBase + c * 16u;
            unsigned go = gByte   + c * 16u;
            asm volatile("global_load_async_to_lds_b128 %0, %1, %2"
                         :: "v"(lo), "v"(go), "s"(nf) : "memory");
        }
        asm volatile("s_wait_asynccnt 0x0" ::: "memory");
    } else if (n < N) {
        // Tail block: plain vector loads into the same LDS row (rows are 16B aligned).
        float4*       row4 = (float4*)(&tile[threadIdx.x * NF]);
        const float4* src4 = (const float4*)(nf + (size_t)n * NF);
#pragma unroll
        for (int c = 0; c < 13; ++c) row4[c] = src4[c];
    }

    if (n < N) {
        const float* row = &tile[threadIdx.x * NF];
        // atoms = argmax over [0,43)   (first-max tie rule -> strict >)
        float best = row[0]; int atom = 0;
#pragma unroll
        for (int k = 1; k < 43; ++k) { float x = row[k]; if (x > best) { best = x; atom = k; } }
        // charge index = argmax over [43,50)   (charge = idx - 3)
        float cb = row[43]; int ci = 0;
#pragma unroll
        for (int k = 1; k < 7; ++k) { float x = row[43 + k]; if (x > cb) { cb = x; ci = k; } }
        // aromatic = argmax over last two cols
        int arom = (row[51] > row[50]) ? 1 : 0;
        int mb   = (int)mbt[atom];   // table holds small exact integers
        meta[n]  = atom | (ci << 8) | (arom << 16) | (mb << 20);
        atomicAdd(&gcount[gid[n]], 1.0f);
    }
}

// ---------------------------------------------------------------------------
// Edge pass: stage 256 edge rows (5 f32 each) into LDS via async global->LDS
// (per wave: 32 rows = 640B = 40 x b128 chunks), then bond-type argmax and
// scatter-add. The three integer-valued sums (bond total, aromatic count,
// degree) are packed into ONE u64 atomic (fields bounded << 2^20), the edge
// logprob sum is a f32 atomic: 2 atomics/edge instead of 4, all L2-resident.
// ---------------------------------------------------------------------------
__global__ void k_edge(const float* __restrict__ ef,
                       const float* __restrict__ elp,
                       const int*   __restrict__ src,
                       const int*   __restrict__ dst,
                       const int*   __restrict__ isgen,
                       const int*   __restrict__ meta,
                       unsigned long long* __restrict__ packAcc,
                       float*       __restrict__ lpAcc,
                       int E) {
    __shared__ __align__(16) float et[TB * 5];      // 5 KB
    const int  lane       = threadIdx.x & 31;
    const int  wave       = threadIdx.x >> 5;
    const int  blockStart = blockIdx.x * TB;
    const int  e          = blockStart + threadIdx.x;
    const bool full       = (blockStart + TB) <= E;

    if (full) {
        unsigned ldsBase = (unsigned)(uintptr_t)(&et[wave * 32 * 5]);
        unsigned gByte   = (unsigned)(blockStart + wave * 32) * 20u;
        {
            unsigned c = (unsigned)lane;
            asm volatile("global_load_async_to_lds_b128 %0, %1, %2"
                         :: "v"(ldsBase + c * 16u), "v"(gByte + c * 16u), "s"(ef) : "memory");
        }
        if (lane < 8) {
            unsigned c = 32u + (unsigned)lane;
            asm volatile("global_load_async_to_lds_b128 %0, %1, %2"
                         :: "v"(ldsBase + c * 16u), "v"(gByte + c * 16u), "s"(ef) : "memory");
        }
        asm volatile("s_wait_asynccnt 0x0" ::: "memory");
    } else if (e < E) {
        float*       row = &et[threadIdx.x * 5];
        const float* r   = ef + (size_t)e * 5;
#pragma unroll
        for (int k = 0; k < 5; ++k) row[k] = r[k];
    }
    if (e >= E) return;

    const float* r = &et[threadIdx.x * 5];   // stride 5: gcd(5,64)=1 -> conflict-free
    float b = r[0]; int bt = 0;
#pragma unroll
    for (int k = 1; k < 5; ++k) { float x = r[k]; if (x > b) { b = x; bt = k; } }

    const int s  = src[e], d = dst[e];
    const int sa = meta[s] & 0xFF;
    const int da = meta[d] & 0xFF;

    const unsigned long long isar =
        (sa != DUMMY && bt == 4 && da != DUMMY) ? 1ull : 0ull;
    int b2 = (bt == 4) ? 1 : bt;          // renumber aromatic
    if (sa == DUMMY)    b2 = 0;
    if (isgen[e] == -1) b2 = 0;

    const unsigned long long pk =
        (unsigned long long)b2 | (isar << 20) | (1ull << 40);  // bonds | aromcnt | deg
    atomicAdd(&packAcc[d], pk);
    atomicAdd(&lpAcc[d], elp[e]);
}

// ---------------------------------------------------------------------------
// Node reward/loss, scatter-add into per-graph sums.
// ---------------------------------------------------------------------------
__global__ void k_node_reward(const int*   __restrict__ meta,
                              const unsigned long long* __restrict__ packAcc,
                              const float* __restrict__ lpAcc,
                              const float* __restrict__ nlp,
                              const int*   __restrict__ gid,
                              const float* __restrict__ gcount,
                              float*       __restrict__ gloss,
                              float*       __restrict__ greward,
                              int N) {
    const int n = blockIdx.x * blockDim.x + threadIdx.x;
    if (n >= N) return;
    const int   m      = meta[n];
    const int   atom   = m & 0xFF;
    const float charge = (float)(((m >> 8) & 0xF) - 3);
    const int   arom   = (m >> 16) & 1;
    const float maxb   = (float)((m >> 20) & 0xF);

    const unsigned long long pk = packAcc[n];
    const unsigned bonds  = (unsigned)(pk & 0xFFFFFu);          // seg-sum of bt2 (exact int)
    const unsigned isarc  = (unsigned)((pk >> 20) & 0xFFFFFu);  // seg-sum of aromatic flags
    const unsigned deg    = (unsigned)(pk >> 40);               // in-degree

    const bool  dummy     = (atom == DUMMY);
    const float num_bonds = (float)bonds;

    const float af = ((isarc > 0) != (arom > 0))      ? 1.0f : 0.0f;
    const float vf = ((num_bonds - charge) > maxb)    ? 1.0f : 0.0f;
    const float bf = (dummy != (bonds == 0))          ? 1.0f : 0.0f;
    float reward = -1.0f * af - 2.0f * vf - 3.0f * bf;

    const int   g   = gid[n];
    const float cnt = gcount[g];
    if (cnt == 1.0f && dummy) reward = -4.0f;

    const float mean_lp = (deg > 0) ? (lpAcc[n] / (float)deg) : 0.0f;
    const float loss    = -(nlp[n] + mean_lp) * reward;

    atomicAdd(&gloss[g],   loss);
    atomicAdd(&greward[g], reward);
}

// ---------------------------------------------------------------------------
// Zero workspace: packed u64 accumulators (N), lp accumulators (N floats) and
// per-graph arrays (3 x B floats). B read from device (num_graphs is a
// 1-element device array; no sync copies -> graph-capture safe).
// ---------------------------------------------------------------------------
__global__ void k_zero(unsigned long long* __restrict__ packAcc,
                       float* __restrict__ lpAcc,
                       float* __restrict__ gl,
                       float* __restrict__ gr,
                       float* __restrict__ gc,
                       const int* __restrict__ nb,
                       int N) {
    const long stride = (long)gridDim.x * blockDim.x;
    const long i0     = (long)blockIdx.x * blockDim.x + threadIdx.x;
    for (long i = i0; i < N; i += stride) { packAcc[i] = 0ull; lpAcc[i] = 0.0f; }
    const int B = *nb;
    for (long i = i0; i < B; i += stride) { gl[i] = 0.0f; gr[i] = 0.0f; gc[i] = 0.0f; }
}

// ---------------------------------------------------------------------------
// Final mean over graphs: single block, wave32 shuffle reduction.
// ---------------------------------------------------------------------------
__global__ void k_final(const float* __restrict__ gloss,
                        const float* __restrict__ greward,
                        const float* __restrict__ gcount,
                        const int*   __restrict__ nb,
                        float*       __restrict__ out) {
    const int B = *nb;
    float sl = 0.0f, sr = 0.0f;
    for (int g = threadIdx.x; g < B; g += blockDim.x) {
        const float c   = gcount[g];
        const float inv = (c > 0.0f) ? (1.0f / c) : 0.0f;
        sl += gloss[g]   * inv;
        sr += greward[g] * inv;
    }
#pragma unroll
    for (int off = 16; off > 0; off >>= 1) {
        sl += __shfl_down(sl, off, 32);
        sr += __shfl_down(sr, off, 32);
    }
    __shared__ float red[64];
    const int lane = threadIdx.x & 31;
    const int w    = threadIdx.x >> 5;
    if (lane == 0) { red[w] = sl; red[32 + w] = sr; }
    __syncthreads();
    if (threadIdx.x == 0) {
        float tl = 0.0f, tr = 0.0f;
        const int nw = blockDim.x >> 5;
        for (int i = 0; i < nw; ++i) { tl += red[i]; tr += red[32 + i]; }
        const float invB = (B > 0) ? (1.0f / (float)B) : 0.0f;
        out[0] = tl * invB;
        out[1] = tr * invB;
    }
}

// ---------------------------------------------------------------------------
// Launch
// ---------------------------------------------------------------------------
extern "C" void kernel_launch(void* const* d_in, const int* in_sizes, int n_in,
                              void* d_out, int out_size, void* d_ws, size_t ws_size,
                              hipStream_t stream) {
    const float* nf  = (const float*)d_in[0];   // node_feats   [N,52]
    const float* ef  = (const float*)d_in[1];   // edge_feats   [E,5]
    const float* nlp = (const float*)d_in[2];   // node_logprobs[N]
    const float* elp = (const float*)d_in[3];   // edge_logprobs[E]
    const float* mbt = (const float*)d_in[4];   // max_bonds_table[43]
    const int*   src = (const int*)d_in[5];
    const int*   dst = (const int*)d_in[6];
    const int*   isg = (const int*)d_in[7];
    const int*   gid = (const int*)d_in[8];
    const int*   nb  = (const int*)d_in[9];     // num_graphs (device scalar)

    const int N = in_sizes[2];
    const int E = in_sizes[3];

    // Workspace layout (float units):
    //   meta[N] | packAcc u64[N] (=2N floats) | lpAcc[N] | gloss[Ncap] | grew[Ncap] | gcnt[Ncap]
    float*              ws   = (float*)d_ws;
    int*                meta = (int*)ws;                              // N ints
    unsigned long long* pAcc = (unsigned long long*)(ws + (size_t)N); // N u64 (8B aligned)
    float*              lpA  = ws + (size_t)3 * N;
    float*              gl   = ws + (size_t)4 * N;
    float*              gr   = ws + (size_t)5 * N;
    float*              gc   = ws + (size_t)6 * N;

    k_zero<<<1024, 256, 0, stream>>>(pAcc, lpA, gl, gr, gc, nb, N);
    k_node_prep<<<(N + TB - 1) / TB, TB, 0, stream>>>(nf, mbt, gid, meta, gc, N);
    k_edge<<<(E + TB - 1) / TB, TB, 0, stream>>>(ef, elp, src, dst, isg, meta, pAcc, lpA, E);
    k_node_reward<<<(N + 255) / 256, 256, 0, stream>>>(meta, pAcc, lpA, nlp, gid, gc, gl, gr, N);
    k_final<<<1, 256, 0, stream>>>(gl, gr, gc, nb, (float*)d_out);
}